// Nested_MNL_2929167696668
// MI455X (gfx1250) — compile-verified
//
#include <hip/hip_runtime.h>
#include <hip/hip_bf16.h>
#include <math.h>

#define BATCH_N   512
#define SEQ_LEN   128
#define NUM_NESTS 500
#define THREADS   256

// ---- CDNA5 async global->LDS copy (ASYNCcnt path) -------------------------
typedef __attribute__((address_space(3))) int  lds_i32_t;
typedef __attribute__((address_space(1))) int  glb_i32_t;   // builtin wants non-const

__device__ __forceinline__ void async_copy_i32(const int* gsrc, int* ldst) {
#if __has_builtin(__builtin_amdgcn_global_load_async_to_lds_b32)
    __builtin_amdgcn_global_load_async_to_lds_b32((glb_i32_t*)gsrc, (lds_i32_t*)ldst, 0, 0);
#else
    *ldst = *gsrc;   // fallback: synchronous load + LDS store
#endif
}

__device__ __forceinline__ void async_wait0() {
#if __has_builtin(__builtin_amdgcn_global_load_async_to_lds_b32)
 #if __has_builtin(__builtin_amdgcn_s_wait_asynccnt)
    __builtin_amdgcn_s_wait_asynccnt(0);
 #else
    asm volatile("s_wait_asynccnt 0" ::: "memory");
 #endif
#endif
}

// ---- CDNA5 matrix-unit wave reduction -------------------------------------
// Full 32-lane sum via V_WMMA_F32_16X16X4_F32 with B = ones:
// D[i][j] = rowsum_A(i); per-lane sum of the 8 D VGPRs yields rows 0-7
// (lanes 0-15) / rows 8-15 (lanes 16-31); one xor-16 shuffle finishes.
// With B all-ones the result equals the sum of ALL A elements regardless of
// the exact lane->(i,k) mapping, so this is layout-robust.
typedef __attribute__((ext_vector_type(2))) float v2f;
typedef __attribute__((ext_vector_type(8))) float v8f;

__device__ __forceinline__ float wave_sum32(float part) {
#if __has_builtin(__builtin_amdgcn_wmma_f32_16x16x4_f32)
    v2f a;     a.x = part; a.y = 0.0f;
    v2f bones; bones.x = 1.0f; bones.y = 1.0f;
    v8f c = {};
    v8f d = __builtin_amdgcn_wmma_f32_16x16x4_f32(
        /*neg_a=*/false, a, /*neg_b=*/false, bones,
        /*c_mod=*/(short)0, c, /*reuse_a=*/false, /*reuse_b=*/false);
    float s = d[0] + d[1] + d[2] + d[3] + d[4] + d[5] + d[6] + d[7];
    s += __shfl_xor(s, 16, 32);
    return s;
#else
    #pragma unroll
    for (int off = 16; off > 0; off >>= 1) part += __shfl_xor(part, off, 32);
    return part;
#endif
}

// ---------------------------------------------------------------------------
// One workgroup per batch row. 256 threads = 8 wave32s.
// ---------------------------------------------------------------------------
__global__ __launch_bounds__(THREADS, 4)
void nested_mnl_kernel(const int*   __restrict__ x,          // (B, S, 2) int32
                       const float* __restrict__ logits,     // (NI+1,)
                       const float* __restrict__ scale_param,// (NN,)
                       const int*   __restrict__ nest_ids,   // (NI,)
                       float*       __restrict__ out)        // (B, S)
{
    __shared__ int   items2[2 * SEQ_LEN];   // raw x row (item stored at even idx)
    __shared__ float nest_sum[NUM_NESTS];
    __shared__ float sel_v[SEQ_LEN];        // V_i / scale_i per slot
    __shared__ float sel_sc[SEQ_LEN];       // scale of slot's nest
    __shared__ int   sel_n[SEQ_LEN];        // nest id per slot
    __shared__ float total_sh;

    const int b   = blockIdx.x;
    const int tid = threadIdx.x;

    // ---- zero accumulators ----
    for (int n = tid; n < NUM_NESTS; n += THREADS) nest_sum[n] = 0.0f;
    if (tid == 0) total_sh = 0.0f;

    // ---- stage the x row into LDS asynchronously (256 contiguous ints) ----
    const int* row = x + (size_t)b * (2 * SEQ_LEN);
    async_copy_i32(row + tid, &items2[tid]);
    async_wait0();                // each wave drains its own ASYNCcnt
    __syncthreads();

    // ---- dedup + per-item weight, scatter into per-nest LDS sums ----
    if (tid < SEQ_LEN) {
        const int id = items2[2 * tid];
        bool first = true;                       // C.at[..].set(1): count once
        for (int j = 0; j < tid; ++j) {
            if (items2[2 * j] == id) { first = false; break; }
        }
        const int   n  = nest_ids[id];           // L2-resident gather
        const float sc = scale_param[n];
        const float v  = logits[id] / sc;        // V_i / scale_i
        sel_v[tid]  = v;
        sel_sc[tid] = sc;
        sel_n[tid]  = n;
        if (first) atomicAdd(&nest_sum[n], __expf(v));   // ds_add_f32
    }
    __syncthreads();

    // ---- total = sum_n (ns==0 ? 1 : ns^scale_n) ----
    float part = 0.0f;
    for (int n = tid; n < NUM_NESTS; n += THREADS) {
        const float ns = nest_sum[n];
        part += (ns == 0.0f) ? 1.0f : __expf(scale_param[n] * __logf(ns));
    }
    // EXEC is all-1s here (divergent loop above has converged): WMMA-legal.
    const float wsum = wave_sum32(part);
    if ((tid & 31) == 0) atomicAdd(&total_sh, wsum);
    __syncthreads();

    // ---- closed-form logP ----
    if (tid < SEQ_LEN) {
        const int   n   = sel_n[tid];
        const float ns  = nest_sum[n];           // > 0: this slot's item is in nest n
        const float sc  = sel_sc[tid];
        const float lp  = sel_v[tid] + (sc - 1.0f) * __logf(ns) - __logf(total_sh);
        out[(size_t)b * SEQ_LEN + tid] = lp;
    }
}

extern "C" void kernel_launch(void* const* d_in, const int* in_sizes, int n_in,
                              void* d_out, int out_size, void* d_ws, size_t ws_size,
                              hipStream_t stream) {
    (void)in_sizes; (void)n_in; (void)out_size; (void)d_ws; (void)ws_size;
    const int*   x      = (const int*)d_in[0];   // (512,128,2)
    /* d_in[1] = x_extra : unused by reference output */
    const float* logits = (const float*)d_in[2]; // (50001,)
    const float* scale  = (const float*)d_in[3]; // (500,)
    const int*   nids   = (const int*)d_in[4];   // (50000,)
    float*       out    = (float*)d_out;         // (512,128)

    nested_mnl_kernel<<<BATCH_N, THREADS, 0, stream>>>(x, logits, scale, nids, out);
}